// IIR2DResidual_11089605558742
// MI455X (gfx1250) — compile-verified
//
#include <hip/hip_runtime.h>
#include <math.h>

// CDNA5 / gfx1250: wave32, WMMA f32 16x16x4.
typedef __attribute__((ext_vector_type(2))) float v2f;
typedef __attribute__((ext_vector_type(8))) float v8f;

#define N_AX   512     // scan axis length (H and W)
#define NTILE  32      // 512 / 16 row-tiles

// ---------------------------------------------------------------------------
// Build the 512x512 combined forward+backward IIR operator M numerically.
// Column j of M is the exact double-scan response to basis vector e_j
// (a = b = 0.5, mirror border), reproducing the reference recurrences:
//   forward closed form (exact powers of two), backward recurrence explicit.
// ---------------------------------------------------------------------------
__global__ void build_M_kernel(float* __restrict__ M) {
  int j = blockIdx.x * blockDim.x + threadIdx.x;
  if (j >= N_AX) return;
  float yprev = 0.0f;
  for (int t = N_AX - 1; t >= 0; --t) {
    float yf;
    if (j == 0)  yf = exp2f((float)(-t));                       // a^t  (c0 = x[0])
    else         yf = (t >= j) ? 0.5f * exp2f((float)(j - t))   // b*a^(t-j)
                               : 0.0f;
    float yb = (t == N_AX - 1) ? yf : 0.5f * yf + 0.5f * yprev; // yb[t]=b*yf+a*yb[t+1]
    yprev = yb;
    M[(size_t)t * N_AX + j] = yb;
  }
}

// B-operand tile: 16 K-rows x 16 columns of the input, as 4 WMMA B operands.
// B layout (16x16x4 f32): VGPR0 lanes0-15 = K=kk*4+0, lanes16-31 = K=kk*4+2;
// VGPR1 = +1. Column index = lane&15.
struct BTile { v2f b[4]; };

__device__ __forceinline__ BTile load_btile(const float* __restrict__ in,
                                            long F, int j, long col, int kh) {
  BTile t;
  if (j >= 0 && j < NTILE) {
    long base = (long)j * 16;
#pragma unroll
    for (int kk = 0; kk < 4; ++kk) {
      const float* p = in + (base + kk * 4 + kh) * F + col;
      t.b[kk].x = p[0];
      t.b[kk].y = p[F];
    }
  } else {
#pragma unroll
    for (int kk = 0; kk < 4; ++kk) { t.b[kk].x = 0.0f; t.b[kk].y = 0.0f; }
  }
  return t;
}

// ---------------------------------------------------------------------------
// Banded operator application: Out[b][t][f] = sum_k M[t][k] * In[b][k][f]
// with the band truncated to k-tiles i-2..i+2 (|t-k|>=33 terms < 4e-10,
// far below f32 rounding noise). One wave per 16-wide column strip,
// marching all 32 row-tiles with a 5-tile register sliding window.
// doMix!=0: fused residual epilogue out = x + sigmoid(alpha[c])*(z - x).
// ---------------------------------------------------------------------------
__global__ void band_apply_kernel(const float* __restrict__ in,
                                  float* __restrict__ out,
                                  const float* __restrict__ M,
                                  const float* __restrict__ xres,
                                  const float* __restrict__ alpha,
                                  int fTilesLog2, int doMix) {
  const int lane   = threadIdx.x & 31;
  const int lane15 = lane & 15;
  const int kh     = (lane < 16) ? 0 : 2;       // K half-select for A/B operands
  const int wid    = blockIdx.x * (blockDim.x >> 5) + (threadIdx.x >> 5);

  const int  ft    = wid & ((1 << fTilesLog2) - 1);
  const int  bslab = wid >> fTilesLog2;
  const long F     = 16L << fTilesLog2;         // fast-dim length (64 or 32768)
  const long col   = (long)ft * 16 + lane15;

  const float* inb  = in  + (long)bslab * N_AX * F;
  float*       outb = out + (long)bslab * N_AX * F;
  const float* xb   = xres ? (xres + (long)bslab * N_AX * F) : nullptr;

  float mix = 0.0f;
  if (doMix) {
    int c = (int)(col & 63);                    // NHWC: channel = col % 64
    mix = 1.0f / (1.0f + expf(-alpha[c]));
  }

  // Prime sliding window for i = 0: tiles j = -2..2 (OOB -> zero).
  BTile win[5];
  win[0] = load_btile(inb, F, -2, col, kh);
  win[1] = load_btile(inb, F, -1, col, kh);
  win[2] = load_btile(inb, F,  0, col, kh);
  win[3] = load_btile(inb, F,  1, col, kh);
  win[4] = load_btile(inb, F,  2, col, kh);

  for (int i = 0; i < NTILE; ++i) {
    v8f acc = {0.f, 0.f, 0.f, 0.f, 0.f, 0.f, 0.f, 0.f};
    const long   t0   = (long)i * 16;
    const float* Mrow = M + (t0 + lane15) * N_AX;   // A row = t0 + (lane&15)

    // Prefetch next window tile's rows (CDNA5 global_prefetch_b8).
    if (i + 3 < NTILE)
      __builtin_prefetch(inb + (long)(i + 3) * 16 * F + col, 0, 1);

#pragma unroll
    for (int jj = 0; jj < 5; ++jj) {
      int j = i - 2 + jj;                        // wave-uniform validity branch:
      if (j >= 0 && j < NTILE) {                 // EXEC stays all-ones for WMMA
#pragma unroll
        for (int kk = 0; kk < 4; ++kk) {
          v2f a;                                 // A: 16x4 slice of M band
          long kc = (long)j * 16 + kk * 4 + kh;
          a.x = Mrow[kc];
          a.y = Mrow[kc + 1];
          acc = __builtin_amdgcn_wmma_f32_16x16x4_f32(
                    /*neg_a=*/false, a, /*neg_b=*/false, win[jj].b[kk],
                    /*c_mod=*/(short)0, acc, /*reuse_a=*/false, /*reuse_b=*/false);
        }
      }
    }

    // D layout: VGPR v -> row t0 + v (lanes 0-15) / t0 + v + 8 (lanes 16-31).
    const long tbase = t0 + ((lane < 16) ? 0 : 8);
#pragma unroll
    for (int v = 0; v < 8; ++v) {
      long idx = (tbase + v) * F + col;
      float z = acc[v];
      if (doMix) {
        float xv = xb[idx];
        outb[idx] = xv + mix * (z - xv);
      } else {
        outb[idx] = z;
      }
    }

    // Rotate window, fetch tile j = i+3.
    win[0] = win[1]; win[1] = win[2]; win[2] = win[3]; win[3] = win[4];
    win[4] = load_btile(inb, F, i + 3, col, kh);
  }
}

// ---------------------------------------------------------------------------
// Launch: build M (1 MB of d_ws), W-axis pass x -> mid (d_ws + 1 MB),
// H-axis pass + fused residual mix: mid -> d_out.
// Workspace needed: 512*512*4 + 4*512*512*64*4 bytes (~269.5 MB).
// ---------------------------------------------------------------------------
extern "C" void kernel_launch(void* const* d_in, const int* in_sizes, int n_in,
                              void* d_out, int out_size, void* d_ws, size_t ws_size,
                              hipStream_t stream) {
  const float* x     = (const float*)d_in[0];   // (4,512,512,64) f32
  const float* alpha = (const float*)d_in[1];   // (64,) f32
  float* out = (float*)d_out;

  float* M   = (float*)d_ws;                                            // 1 MB
  float* mid = (float*)((char*)d_ws + (size_t)N_AX * N_AX * sizeof(float));

  // Operator matrix (recomputed every call: deterministic, cheap).
  build_M_kernel<<<2, 256, 0, stream>>>(M);

  // Pass 1 — along W: slabs = N*H = 2048, F = 64 (fTilesLog2 = 2).
  // waves = 2048*4 = 8192 -> 1024 blocks of 256 threads (8 waves each).
  band_apply_kernel<<<1024, 256, 0, stream>>>(x, mid, M, nullptr, nullptr, 2, 0);

  // Pass 2 — along H: slabs = N = 4, F = W*C = 32768 (fTilesLog2 = 11),
  // fused per-channel sigmoid mix. waves = 4*2048 = 8192 -> 1024 blocks.
  band_apply_kernel<<<1024, 256, 0, stream>>>(mid, out, M, x, alpha, 11, 1);
}